// PointNetFeatureExtractor_88424786690346
// MI455X (gfx1250) — compile-verified
//
#include <hip/hip_runtime.h>
#include <hip/hip_bf16.h>
#include <stdint.h>

// ---------------------------------------------------------------------------
// PointNet feature extractor for MI455X (gfx1250), all GEMMs on f32 WMMA
// (V_WMMA_F32_16X16X4_F32), scatter-max via ordered-key global_atomic_max_u32.
// ---------------------------------------------------------------------------

typedef float v2f __attribute__((ext_vector_type(2)));
typedef float v8f __attribute__((ext_vector_type(8)));

#define LN_EPS 1e-5f

__device__ __forceinline__ v8f wmma_f32_16x16x4(v2f a, v2f b, v8f c) {
  // 8 args: (neg_a, A, neg_b, B, c_mod, C, reuse_a, reuse_b)
  return __builtin_amdgcn_wmma_f32_16x16x4_f32(false, a, false, b, (short)0, c,
                                               false, false);
}

// Butterfly add across each 16-lane half of a wave32 (offsets 1,2,4,8 never
// cross the half boundary) -> per-row LayerNorm reductions in WMMA C layout.
__device__ __forceinline__ float redux16_add(float v) {
  v += __shfl_xor(v, 1, 32);
  v += __shfl_xor(v, 2, 32);
  v += __shfl_xor(v, 4, 32);
  v += __shfl_xor(v, 8, 32);
  return v;
}

// Monotone float -> u32 key so unsigned atomic max == float max.
// Key 0 decodes from a NaN bit pattern (never produced) => empty sentinel.
__device__ __forceinline__ unsigned f2ord(float f) {
  unsigned u = __float_as_uint(f);
  return (u & 0x80000000u) ? ~u : (u | 0x80000000u);
}
__device__ __forceinline__ float ord2f_or_zero(unsigned k) {
  if (k == 0u) return 0.0f;  // empty segment -> 0 (reference semantics)
  return (k & 0x80000000u) ? __uint_as_float(k ^ 0x80000000u)
                           : __uint_as_float(~k);
}

// ---------------------------------------------------------------------------
// Stage 1: per-point MLP1 + scatter-max.  8 waves/block, 16 points per wave.
//   t1 = [x | 1] @ [W1 ; b1]      (K=4 WMMA, bias folded into padded K slot)
//   h  = relu(LN64(t1)) @ W2 + b2 (16x K=4 WMMA chain per 16-col tile)
//   segKeys[bidx[p]][:] = max(..., ordkey(h))
// ---------------------------------------------------------------------------
__global__ __launch_bounds__(256) void pointnet_stage1(
    const float* __restrict__ x, const long long* __restrict__ bidx,
    const float* __restrict__ W1, const float* __restrict__ b1,
    const float* __restrict__ g1, const float* __restrict__ be1,
    const float* __restrict__ W2, const float* __restrict__ b2,
    unsigned* __restrict__ segKeys, int P) {
  __shared__ __align__(16) float w2lds[64 * 64];     // 16 KB: staged W2
  __shared__ __align__(16) float hbuf[8 * 16 * 66];  // 33 KB: C->A relayout

  const int tid = threadIdx.x;
  const int lane = tid & 31;
  const int wv = tid >> 5;
  const int half = lane >> 4;  // 0: K 0..1 / rows 0..7 ; 1: K 2..3 / rows 8..15
  const int l16 = lane & 15;

  for (int i = tid; i < 64 * 64; i += 256) w2lds[i] = W2[i];

  const int p0 = (blockIdx.x * 8 + wv) * 16;
  if (p0 + 16 > P) { __syncthreads(); return; }

  // ---- GEMM1: A[16x4] = [x0 x1 x2 1], B[4x64] = [W1 ; b1]
  v2f a1;
  {
    const float* xp = x + (size_t)(p0 + l16) * 3;
    if (half == 0) { a1.x = xp[0]; a1.y = xp[1]; }
    else           { a1.x = xp[2]; a1.y = 1.0f; }
  }

  const v8f vzero = {0.f, 0.f, 0.f, 0.f, 0.f, 0.f, 0.f, 0.f};
  v8f c[4];
#pragma unroll
  for (int t = 0; t < 4; ++t) {
    const int n = 16 * t + l16;
    v2f b;
    b.x = half ? W1[2 * 64 + n] : W1[0 * 64 + n];
    b.y = half ? b1[n]          : W1[1 * 64 + n];
    c[t] = wmma_f32_16x16x4(a1, b, vzero);
  }

  float gl[4], bl[4];
#pragma unroll
  for (int t = 0; t < 4; ++t) {
    gl[t] = g1[16 * t + l16];
    bl[t] = be1[16 * t + l16];
  }

  // ---- LayerNorm(64) + ReLU, write A-layout staging buffer (stride 66)
  float* hb = hbuf + wv * (16 * 66);
#pragma unroll
  for (int v = 0; v < 8; ++v) {
    float s = 0.f, q = 0.f;
#pragma unroll
    for (int t = 0; t < 4; ++t) { float e = c[t][v]; s += e; q += e * e; }
    s = redux16_add(s);
    q = redux16_add(q);
    const float mu = s * (1.0f / 64.0f);
    const float var = q * (1.0f / 64.0f) - mu * mu;
    const float rstd = rsqrtf(var + LN_EPS);
    const int m = v + 8 * half;
#pragma unroll
    for (int t = 0; t < 4; ++t) {
      float val = (c[t][v] - mu) * rstd * gl[t] + bl[t];
      hb[m * 66 + 16 * t + l16] = fmaxf(val, 0.f);
    }
  }
  __syncthreads();

  // ---- GEMM2: h[16x64] @ W2[64x64] as 16 K-chunks of 4
  v8f d[4];
#pragma unroll
  for (int t = 0; t < 4; ++t) d[t] = vzero;

  for (int kk = 0; kk < 16; ++kk) {
    const int k0 = 4 * kk + 2 * half;
    const v2f a = *(const v2f*)(hb + l16 * 66 + k0);  // 8B aligned (66*m+k0 even)
#pragma unroll
    for (int t = 0; t < 4; ++t) {
      const int n = 16 * t + l16;
      v2f b;
      b.x = w2lds[k0 * 64 + n];
      b.y = w2lds[(k0 + 1) * 64 + n];
      d[t] = wmma_f32_16x16x4(a, b, d[t]);
    }
  }

  float b2l[4];
#pragma unroll
  for (int t = 0; t < 4; ++t) b2l[t] = b2[16 * t + l16];

  // ---- + b2, scatter-max into L2-resident key table
#pragma unroll
  for (int v = 0; v < 8; ++v) {
    const int m = v + 8 * half;
    const long long sid = bidx[p0 + m];
    unsigned* seg = segKeys + (size_t)sid * 64;
#pragma unroll
    for (int t = 0; t < 4; ++t) {
      const float val = d[t][v] + b2l[t];
      atomicMax(&seg[16 * t + l16], f2ord(val));
    }
  }
}

// ---------------------------------------------------------------------------
// Stage 2: per-node MLP2.  4 waves/block, 16 nodes per wave.
//   t2  = seg[16x64] @ W3[64x128] + b3
//   out = relu(LN128(t2)) @ W4[128x128] + b4
// ---------------------------------------------------------------------------
__global__ __launch_bounds__(128) void pointnet_stage2(
    const unsigned* __restrict__ segKeys,
    const float* __restrict__ W3, const float* __restrict__ b3,
    const float* __restrict__ g2, const float* __restrict__ be2,
    const float* __restrict__ W4, const float* __restrict__ b4,
    float* __restrict__ out, int N) {
  __shared__ __align__(16) float hbuf[4 * 16 * 130];  // 33 KB staging

  const int tid = threadIdx.x;
  const int lane = tid & 31;
  const int wv = tid >> 5;
  const int half = lane >> 4;
  const int l16 = lane & 15;

  const int n0 = (blockIdx.x * 4 + wv) * 16;
  if (n0 + 16 > N) { __syncthreads(); return; }

  const v8f vzero = {0.f, 0.f, 0.f, 0.f, 0.f, 0.f, 0.f, 0.f};

  // ---- GEMM1: decode pooled keys straight into A-layout, K=64
  v8f c[8];
#pragma unroll
  for (int t = 0; t < 8; ++t) c[t] = vzero;

  const unsigned* row = segKeys + (size_t)(n0 + l16) * 64;
  for (int kk = 0; kk < 16; ++kk) {
    const int k0 = 4 * kk + 2 * half;
    v2f a;
    a.x = ord2f_or_zero(row[k0]);      // aligned u32 pair -> b64 load
    a.y = ord2f_or_zero(row[k0 + 1]);
#pragma unroll
    for (int t = 0; t < 8; ++t) {
      const int n = 16 * t + l16;
      v2f b;
      b.x = W3[k0 * 128 + n];
      b.y = W3[(k0 + 1) * 128 + n];
      c[t] = wmma_f32_16x16x4(a, b, c[t]);
    }
  }

  float b3l[8], gl[8], bl[8];
#pragma unroll
  for (int t = 0; t < 8; ++t) {
    b3l[t] = b3[16 * t + l16];
    gl[t] = g2[16 * t + l16];
    bl[t] = be2[16 * t + l16];
  }

  // ---- + b3, LayerNorm(128) + ReLU -> LDS (stride 130, keeps v2f aligned)
  float* hb = hbuf + wv * (16 * 130);
#pragma unroll
  for (int v = 0; v < 8; ++v) {
    float s = 0.f, q = 0.f;
#pragma unroll
    for (int t = 0; t < 8; ++t) {
      const float e = c[t][v] + b3l[t];
      s += e;
      q += e * e;
    }
    s = redux16_add(s);
    q = redux16_add(q);
    const float mu = s * (1.0f / 128.0f);
    const float var = q * (1.0f / 128.0f) - mu * mu;
    const float rstd = rsqrtf(var + LN_EPS);
    const int m = v + 8 * half;
#pragma unroll
    for (int t = 0; t < 8; ++t) {
      const float e = c[t][v] + b3l[t];
      const float val = (e - mu) * rstd * gl[t] + bl[t];
      hb[m * 130 + 16 * t + l16] = fmaxf(val, 0.f);
    }
  }
  __syncthreads();

  // ---- GEMM2: [16x128] @ W4[128x128], K=128 as 32 chunks of 4
  v8f d[8];
#pragma unroll
  for (int t = 0; t < 8; ++t) d[t] = vzero;

  for (int kk = 0; kk < 32; ++kk) {
    const int k0 = 4 * kk + 2 * half;
    const v2f a = *(const v2f*)(hb + l16 * 130 + k0);
#pragma unroll
    for (int t = 0; t < 8; ++t) {
      const int n = 16 * t + l16;
      v2f b;
      b.x = W4[k0 * 128 + n];
      b.y = W4[(k0 + 1) * 128 + n];
      d[t] = wmma_f32_16x16x4(a, b, d[t]);
    }
  }

  float b4l[8];
#pragma unroll
  for (int t = 0; t < 8; ++t) b4l[t] = b4[16 * t + l16];

#pragma unroll
  for (int v = 0; v < 8; ++v) {
    const int m = v + 8 * half;
    float* orow = out + (size_t)(n0 + m) * 128;
#pragma unroll
    for (int t = 0; t < 8; ++t) orow[16 * t + l16] = d[t][v] + b4l[t];
  }
}

// ---------------------------------------------------------------------------
extern "C" void kernel_launch(void* const* d_in, const int* in_sizes, int n_in,
                              void* d_out, int out_size, void* d_ws,
                              size_t ws_size, hipStream_t stream) {
  const float* x = (const float*)d_in[0];
  const long long* bidx = (const long long*)d_in[1];  // int64 batch indices
  // d_in[2] = num_nodes scalar lives on device; N derived from out_size.
  const float* W1 = (const float*)d_in[3];
  const float* b1 = (const float*)d_in[4];
  const float* g1 = (const float*)d_in[5];
  const float* be1 = (const float*)d_in[6];
  const float* W2 = (const float*)d_in[7];
  const float* b2 = (const float*)d_in[8];
  const float* W3 = (const float*)d_in[9];
  const float* b3 = (const float*)d_in[10];
  const float* g2 = (const float*)d_in[11];
  const float* be2 = (const float*)d_in[12];
  const float* W4 = (const float*)d_in[13];
  const float* b4 = (const float*)d_in[14];
  float* out = (float*)d_out;

  const int P = in_sizes[0] / 3;    // 2097152
  const int N = out_size / 128;     // 65536

  unsigned* segKeys = (unsigned*)d_ws;  // [N,64] ordered-key max table (16 MB)
  hipMemsetAsync(segKeys, 0, (size_t)N * 64 * sizeof(unsigned), stream);

  const int blocks1 = (P + 127) / 128;  // 8 waves x 16 points
  pointnet_stage1<<<blocks1, 256, 0, stream>>>(x, bidx, W1, b1, g1, be1, W2, b2,
                                               segKeys, P);

  const int blocks2 = (N + 63) / 64;    // 4 waves x 16 nodes
  pointnet_stage2<<<blocks2, 128, 0, stream>>>(segKeys, W3, b3, g2, be2, W4, b4,
                                               out, N);
}